// QuantumConv_42004780155287
// MI455X (gfx1250) — compile-verified
//
#include <hip/hip_runtime.h>
#include <hip/hip_bf16.h>
#include <math.h>

// ---------------------------------------------------------------------------
// Quantum conv layer, restructured as GEMMs for CDNA5 WMMA (gfx1250, wave32).
//
//  K1 conv   : x[4,16,64,64] * w[8,16,3,3] + b -> ang[16384,8]
//  K2 encode : st0[m,i] = prod_w (bit_{7-w}(i) ? sin(a_w/2) : cos(a_w/2))  (REAL)
//  K3 buildB : Ball[t][512][256]: rows 0..255 = Re(U_t), 256..511 = Im(U_t)
//  K4 buildW : W[t][32][256] = fc_w[:, t*8+w] folded with Z-sign patterns
//  K5 gemm   : V = st0 @ Ball[t]^T   (f32 WMMA 16x16x4, 32x32 tile/wave)
//  K6 fold   : P = Vr^2+Vi^2 ; out += P @ W[t]^T ; stage 0 adds fc_b (NCHW).
// ---------------------------------------------------------------------------

#define NPX   16384     // 4*64*64
#define NAMP  256       // 2^8
#define NQ    8

typedef float v2f __attribute__((ext_vector_type(2)));
typedef float v8f __attribute__((ext_vector_type(8)));

struct cplx { float x, y; };
__device__ __forceinline__ cplx cmul(cplx a, cplx b) {
    return { a.x * b.x - a.y * b.y, a.x * b.y + a.y * b.x };
}
__device__ __forceinline__ cplx cadd(cplx a, cplx b) { return { a.x + b.x, a.y + b.y }; }

// ------------------------- K1: conv 3x3 SAME -------------------------------
__global__ void k_conv(const float* __restrict__ x, const float* __restrict__ cw,
                       const float* __restrict__ cb, float* __restrict__ ang) {
    int gid = blockIdx.x * blockDim.x + threadIdx.x;   // NPX*8
    if (gid >= NPX * NQ) return;
    int co = gid & 7;
    int m  = gid >> 3;
    int b  = m >> 12;
    int h  = (m >> 6) & 63;
    int w  = m & 63;
    float acc = cb[co];
    for (int ci = 0; ci < 16; ++ci) {
        for (int kh = 0; kh < 3; ++kh) {
            int hh = h + kh - 1;
            if (hh < 0 || hh >= 64) continue;
            for (int kw = 0; kw < 3; ++kw) {
                int ww = w + kw - 1;
                if (ww < 0 || ww >= 64) continue;
                acc += x[((b * 16 + ci) * 64 + hh) * 64 + ww]
                     * cw[((co * 16 + ci) * 3 + kh) * 3 + kw];
            }
        }
    }
    ang[m * NQ + co] = acc;
}

// ------------------------- K2: RY product-state encode ---------------------
__global__ void k_encode(const float* __restrict__ ang, float* __restrict__ st0) {
    __shared__ float cs[NQ], sn[NQ];
    int m   = blockIdx.x;          // one pixel per block
    int tid = threadIdx.x;         // 256
    if (tid < NQ) {
        float a = 0.5f * ang[m * NQ + tid];
        cs[tid] = cosf(a);
        sn[tid] = sinf(a);
    }
    __syncthreads();
    float v = 1.0f;
    #pragma unroll
    for (int w = 0; w < NQ; ++w)
        v *= ((tid >> (7 - w)) & 1) ? sn[w] : cs[w];
    st0[m * NAMP + tid] = v;
}

// ------------------------- K3: build stage matrices ------------------------
__device__ __forceinline__ void apply1(cplx* st, int b, cplx U00, cplx U01,
                                       cplx U10, cplx U11, int tid) {
    if (tid < 128) {
        int lo = tid & ((1 << b) - 1);
        int i0 = ((tid >> b) << (b + 1)) | lo;
        int i1 = i0 | (1 << b);
        cplx a0 = st[i0], a1 = st[i1];
        st[i0] = cadd(cmul(U00, a0), cmul(U01, a1));
        st[i1] = cadd(cmul(U10, a0), cmul(U11, a1));
    }
}

__device__ __forceinline__ void applyc(cplx* st, int cb, int tb, cplx U00, cplx U01,
                                       cplx U10, cplx U11, int tid) {
    if (tid < 64) {
        int p1 = cb < tb ? cb : tb;
        int p2 = cb < tb ? tb : cb;
        int x = tid;
        x = ((x >> p1) << (p1 + 1)) | (x & ((1 << p1) - 1));
        x = ((x >> p2) << (p2 + 1)) | (x & ((1 << p2) - 1));
        int i0 = x | (1 << cb);              // control bit set, target bit 0
        int i1 = i0 | (1 << tb);
        cplx a0 = st[i0], a1 = st[i1];
        st[i0] = cadd(cmul(U00, a0), cmul(U01, a1));
        st[i1] = cadd(cmul(U10, a0), cmul(U11, a1));
    }
}

__device__ __forceinline__ void u3_of(const float* p, cplx& U00, cplx& U01,
                                      cplx& U10, cplx& U11) {
    float th = p[0], ph = p[1], lam = p[2];
    float c = cosf(0.5f * th), s = sinf(0.5f * th);
    U00 = { c, 0.0f };
    U01 = { -cosf(lam) * s, -sinf(lam) * s };
    U10 = { cosf(ph) * s, sinf(ph) * s };
    U11 = { cosf(ph + lam) * c, sinf(ph + lam) * c };
}

__global__ void k_buildB(const float* __restrict__ u3p, const float* __restrict__ cu3p,
                         float* __restrict__ Ball) {
    __shared__ cplx st[NAMP];
    int t   = blockIdx.x >> 8;     // stage 0..2
    int c   = blockIdx.x & 255;    // column of U_t
    int tid = threadIdx.x;         // 128
    for (int i = tid; i < NAMP; i += 128)
        st[i] = { (i == c) ? 1.0f : 0.0f, 0.0f };
    __syncthreads();

    cplx U00, U01, U10, U11;
    for (int blk = 0; blk < 5; ++blk) {
        for (int w = 0; w < NQ; ++w) {
            u3_of(&u3p[(blk * NQ + w) * 3], U00, U01, U10, U11);
            apply1(st, 7 - w, U00, U01, U10, U11, tid);
            __syncthreads();
        }
        for (int w = 0; w < NQ; ++w) {
            u3_of(&cu3p[(blk * NQ + w) * 3], U00, U01, U10, U11);
            applyc(st, 7 - w, 7 - ((w + 1) & 7), U00, U01, U10, U11, tid);
            __syncthreads();
        }
    }
    const float r = 0.70710678118654752f;
    if (t >= 1) {                                    // H on every wire
        cplx H00 = { r, 0 }, H01 = { r, 0 }, H10 = { r, 0 }, H11 = { -r, 0 };
        for (int w = 0; w < NQ; ++w) { apply1(st, 7 - w, H00, H01, H10, H11, tid); __syncthreads(); }
    }
    if (t >= 2) {                                    // YROT = H @ diag(1,-i) on every wire
        cplx Y00 = { r, 0 }, Y01 = { 0, -r }, Y10 = { r, 0 }, Y11 = { 0, r };
        for (int w = 0; w < NQ; ++w) { apply1(st, 7 - w, Y00, Y01, Y10, Y11, tid); __syncthreads(); }
    }
    for (int i = tid; i < NAMP; i += 128) {
        Ball[(t * 512 + i) * NAMP + c]       = st[i].x;   // real row
        Ball[(t * 512 + 256 + i) * NAMP + c] = st[i].y;   // imag row
    }
}

// ------------------------- K4: folded measurement+FC weights ---------------
__global__ void k_buildW(const float* __restrict__ fc_w, float* __restrict__ W) {
    int idx = blockIdx.x * blockDim.x + threadIdx.x;   // 3*32*256
    if (idx >= 3 * 32 * 256) return;
    int i = idx & 255;
    int o = (idx >> 8) & 31;
    int t = idx >> 13;
    float acc = 0.0f;
    #pragma unroll
    for (int w = 0; w < NQ; ++w) {
        float sgn = ((i >> (7 - w)) & 1) ? -1.0f : 1.0f;
        acc += fc_w[o * 24 + t * NQ + w] * sgn;
    }
    W[idx] = acc;
}

// ------------------------- K5: WMMA GEMM V = st0 @ Bt^T --------------------
// A = st0 [M=16384, K=256]; B = Bt [N=512, K=256] row-major; V [M, 512].
// One wave computes a 32x32 macro-tile (2x2 register blocking): each K-step
// loads 2 A + 2 B fragments and issues 4 chained V_WMMA_F32_16X16X4_F32,
// doubling the WMMA:load ratio vs a single-tile wave.
__global__ void k_gemm_wmma(const float* __restrict__ A, const float* __restrict__ Bm,
                            float* __restrict__ V) {
    const int K = NAMP, Nn = 512;
    int wave = (blockIdx.x * blockDim.x + threadIdx.x) >> 5;
    int lane = threadIdx.x & 31;
    int nt = wave & 15;            // 16 N macro-tiles (512/32)
    int mt = wave >> 4;            // 512 M macro-tiles (16384/32)
    int m0 = mt << 5, n0 = nt << 5;

    int row  = lane & 15;
    int koff = (lane >> 4) << 1;   // lanes 0-15 -> K0/K1, lanes 16-31 -> K2/K3
    const float* Ap0 = A  + (m0 + row) * K + koff;
    const float* Ap1 = Ap0 + 16 * K;
    const float* Bp0 = Bm + (n0 + row) * K + koff;
    const float* Bp1 = Bp0 + 16 * K;

    v8f c00 = { 0, 0, 0, 0, 0, 0, 0, 0 };
    v8f c01 = c00, c10 = c00, c11 = c00;
    for (int k = 0; k < K; k += 4) {
        v2f a0, a1, b0, b1;
        a0[0] = Ap0[k]; a0[1] = Ap0[k + 1];
        a1[0] = Ap1[k]; a1[1] = Ap1[k + 1];
        b0[0] = Bp0[k]; b0[1] = Bp0[k + 1];
        b1[0] = Bp1[k]; b1[1] = Bp1[k + 1];
        c00 = __builtin_amdgcn_wmma_f32_16x16x4_f32(false, a0, false, b0, (short)0, c00, false, false);
        c01 = __builtin_amdgcn_wmma_f32_16x16x4_f32(false, a0, false, b1, (short)0, c01, false, false);
        c10 = __builtin_amdgcn_wmma_f32_16x16x4_f32(false, a1, false, b0, (short)0, c10, false, false);
        c11 = __builtin_amdgcn_wmma_f32_16x16x4_f32(false, a1, false, b1, (short)0, c11, false, false);
    }
    // D layout: lanes 0-15 -> M = base+r, lanes 16-31 -> M = base+8+r; N = base+(lane&15)
    int mr   = (lane >> 4) << 3;
    int ncol = row;
    #pragma unroll
    for (int r = 0; r < 8; ++r) {
        V[(m0      + mr + r) * Nn + n0      + ncol] = c00[r];
        V[(m0      + mr + r) * Nn + n0 + 16 + ncol] = c01[r];
        V[(m0 + 16 + mr + r) * Nn + n0      + ncol] = c10[r];
        V[(m0 + 16 + mr + r) * Nn + n0 + 16 + ncol] = c11[r];
    }
}

// ------------------------- K6: fold |V|^2 through W, scatter NCHW ----------
__global__ void k_fold(const float* __restrict__ V, const float* __restrict__ W,
                       const float* __restrict__ fc_b, float* __restrict__ out, int t) {
    __shared__ float Wl[32 * 257];     // +1 pad per row: kills bank conflicts
    __shared__ float Pl[8 * 264];      // +8 pad: distinct banks across p-groups
    int tid = threadIdx.x;             // 256
    for (int idx = tid; idx < 32 * 256; idx += 256) {
        int o = idx >> 8, k = idx & 255;
        Wl[o * 257 + k] = W[t * 32 * 256 + idx];
    }
    int m0 = blockIdx.x << 3;          // 8 pixels per block
    #pragma unroll
    for (int p = 0; p < 8; ++p) {
        int m = m0 + p;
        float vr = V[m * 512 + tid];
        float vi = V[m * 512 + 256 + tid];
        Pl[p * 264 + tid] = vr * vr + vi * vi;
    }
    __syncthreads();
    int p = tid >> 5, o = tid & 31;
    float acc = 0.0f;
    for (int k = 0; k < 256; ++k)
        acc += Pl[p * 264 + k] * Wl[o * 257 + k];
    int m = m0 + p;
    int b = m >> 12, h = (m >> 6) & 63, w = m & 63;
    int oi = ((b * 32 + o) * 64 + h) * 64 + w;
    if (t == 0) out[oi] = fc_b[o] + acc;
    else        out[oi] += acc;
}

// ---------------------------------------------------------------------------
extern "C" void kernel_launch(void* const* d_in, const int* in_sizes, int n_in,
                              void* d_out, int out_size, void* d_ws, size_t ws_size,
                              hipStream_t stream) {
    const float* x      = (const float*)d_in[0];   // [4,16,64,64]
    const float* conv_w = (const float*)d_in[1];   // [8,16,3,3]
    const float* conv_b = (const float*)d_in[2];   // [8]
    const float* u3p    = (const float*)d_in[3];   // [5,8,3]
    const float* cu3p   = (const float*)d_in[4];   // [5,8,3]
    const float* fc_w   = (const float*)d_in[5];   // [32,24]
    const float* fc_b   = (const float*)d_in[6];   // [32]
    float* out = (float*)d_out;                    // [4,32,64,64] f32

    float* ws   = (float*)d_ws;
    float* ang  = ws;                              // NPX*8          = 131072
    float* st0  = ang  + NPX * NQ;                 // NPX*256        = 4194304
    float* Ball = st0  + NPX * NAMP;               // 3*512*256      = 393216
    float* Wm   = Ball + 3 * 512 * NAMP;           // 3*32*256       = 24576
    float* V    = Wm   + 3 * 32 * NAMP;            // NPX*512        = 8388608
    (void)in_sizes; (void)n_in; (void)out_size; (void)ws_size;

    k_conv  <<<(NPX * NQ + 255) / 256, 256, 0, stream>>>(x, conv_w, conv_b, ang);
    k_encode<<<NPX, 256, 0, stream>>>(ang, st0);
    k_buildB<<<3 * NAMP, 128, 0, stream>>>(u3p, cu3p, Ball);
    k_buildW<<<(3 * 32 * NAMP + 255) / 256, 256, 0, stream>>>(fc_w, Wm);

    for (int t = 0; t < 3; ++t) {
        // 512 M-macrotiles * 16 N-macrotiles = 8192 waves; 8 waves/block -> 1024 blocks
        k_gemm_wmma<<<1024, 256, 0, stream>>>(st0, Ball + t * 512 * NAMP, V);
        k_fold     <<<NPX / 8, 256, 0, stream>>>(V, Wm, fc_b, out, t);
    }
}